// TorchBSplineBasis_37941741092938
// MI455X (gfx1250) — compile-verified
//
#include <hip/hip_runtime.h>

// B-spline tensor-product basis, MI455X (gfx1250, wave32).
// Memory-bound: ~204 MB traffic -> ~8.8 us floor @ 23.3 TB/s.
// Outer product fully in registers with compile-time indices, packed 100KB
// LDS tile, linear LDS->global copy-out via CDNA5
// GLOBAL_STORE_ASYNC_FROM_LDS_B128 (ASYNCcnt path); ds_load+NT-store fallback.

#define NUM_BASIS 10
#define DEGREE    3
#define NKNOTS    14          // NUM_BASIS + DEGREE + 1
#define N0        13          // degree-0 intervals
#define RPB       256         // rows per block
#define OPR       100         // outputs per row (NUM_BASIS^2)
#define CPR       25          // 16B chunks per row

typedef float v4f __attribute__((ext_vector_type(4)));
typedef int   v4i __attribute__((ext_vector_type(4)));  // builtin's pointee type

#if defined(__AMDGCN__) && __has_builtin(__builtin_amdgcn_global_store_async_from_lds_b128)
#define USE_ASYNC 1
#else
#define USE_ASYNC 0
#endif

__global__ __launch_bounds__(RPB) void
bspline_tensor_basis_kernel(const float* __restrict__ t,
                            const float* __restrict__ knots,
                            float* __restrict__ out, int batch)
{
    __shared__ __align__(16) float sTile[RPB * OPR];  // 100 KB packed product tile
    __shared__ __align__(16) float sKn[2 * NKNOTS];
    __shared__ __align__(16) float sRcp[2][40];       // 36 used/dim

    const int tid = threadIdx.x;

    // ---- Stage knots; precompute uniform DP denominator reciprocals ----
    if (tid < 2 * NKNOTS) sKn[tid] = knots[tid];
    __syncthreads();
    if (tid < 72) {
        int d = tid / 36, u = tid % 36;
        int k, j;
        if (u < 13)      { k = 1; j = u;      }
        else if (u < 25) { k = 2; j = u - 13; }
        else             { k = 3; j = u - 25; }
        float dd = sKn[d * NKNOTS + k + j] - sKn[d * NKNOTS + j];
        sRcp[d][u] = (dd == 0.0f) ? 0.0f : 1.0f / dd;
    }
    __syncthreads();

    // ---- Phase 1: per-thread Cox-de Boor DP + outer product, all in VGPRs ----
    const int row0 = blockIdx.x * RPB;
    const int row  = row0 + tid;
    if (row < batch) {
        const float2 tv = ((const float2*)t)[row];   // coalesced b64 load
        const float tvd[2] = { tv.x, tv.y };
        float bas[2][NUM_BASIS];

        #pragma unroll
        for (int d = 0; d < 2; ++d) {
            float kn[NKNOTS];
            #pragma unroll
            for (int j = 0; j < NKNOTS; ++j) kn[j] = sKn[d * NKNOTS + j];
            float rc[36];
            #pragma unroll
            for (int u = 0; u < 36; ++u) rc[u] = sRcp[d][u];

            const float tc = tvd[d];
            float B[N0];
            #pragma unroll
            for (int j = 0; j < N0; ++j)
                B[j] = (kn[j] <= tc && tc < kn[j + 1]) ? 1.0f : 0.0f;

            int base = 0;
            #pragma unroll
            for (int k = 1; k <= DEGREE; ++k) {
                #pragma unroll
                for (int j = 0; j < N0 - k; ++j) {
                    // d2[j] == d1[j+1] -> shared reciprocal table
                    float w1 = (tc - kn[j])         * rc[base + j];
                    float w2 = (kn[k + 1 + j] - tc) * rc[base + j + 1];
                    B[j] = w1 * B[j] + w2 * B[j + 1];
                }
                base += N0 - k + 1;
            }
            #pragma unroll
            for (int j = 0; j < NUM_BASIS; ++j) bas[d][j] = B[j];
        }

        // Outer product: idx/10, idx%10 are COMPILE-TIME after full unroll.
        // Row stride 100 floats = 4*25 (25 odd) -> bank-optimal b128 stores.
        #pragma unroll
        for (int c = 0; c < CPR; ++c) {
            v4f v;
            #pragma unroll
            for (int e = 0; e < 4; ++e) {
                const int idx = c * 4 + e;                  // constant
                v[e] = bas[0][idx / NUM_BASIS] * bas[1][idx % NUM_BASIS];
            }
            *(v4f*)&sTile[tid * OPR + c * 4] = v;           // ds_store_b128
        }
    }
    __syncthreads();

    // ---- Phase 2: linear LDS -> global copy-out, zero index math ----
    float* gout = out + row0 * OPR;      // fits int: 50M floats max
    const int remRows = batch - row0;

    if (remRows >= RPB) {
        #pragma unroll
        for (int s = 0; s < CPR; ++s) {
            const int g = s * RPB + tid;                    // lane-consecutive
#if USE_ASYNC
            __builtin_amdgcn_global_store_async_from_lds_b128(
                (v4i*)(gout + g * 4), (v4i*)(sTile + g * 4), 0, 0);
#else
            v4f v = *(const v4f*)(sTile + g * 4);
            __builtin_nontemporal_store(v, (v4f*)(gout + g * 4));
#endif
        }
    } else {
        // Tail block: rows are packed, so chunk validity is a plain compare.
        const int validChunks = remRows * CPR;
        #pragma unroll
        for (int s = 0; s < CPR; ++s) {
            const int g = s * RPB + tid;
            if (g < validChunks) {
#if USE_ASYNC
                __builtin_amdgcn_global_store_async_from_lds_b128(
                    (v4i*)(gout + g * 4), (v4i*)(sTile + g * 4), 0, 0);
#else
                v4f v = *(const v4f*)(sTile + g * 4);
                __builtin_nontemporal_store(v, (v4f*)(gout + g * 4));
#endif
            }
        }
    }

#if USE_ASYNC
#if __has_builtin(__builtin_amdgcn_s_wait_asynccnt)
    __builtin_amdgcn_s_wait_asynccnt(0);   // S_ENDPGM would also wait-idle
#endif
#endif
}

extern "C" void kernel_launch(void* const* d_in, const int* in_sizes, int n_in,
                              void* d_out, int out_size, void* d_ws, size_t ws_size,
                              hipStream_t stream) {
    const float* t     = (const float*)d_in[0];   // (B, 2) f32
    const float* knots = (const float*)d_in[1];   // (2, 14) f32
    float*       out   = (float*)d_out;           // (B, 100) f32

    const int batch = in_sizes[0] / 2;
    const int grid  = (batch + RPB - 1) / RPB;
    bspline_tensor_basis_kernel<<<grid, RPB, 0, stream>>>(t, knots, out, batch);
}